// GCN_HL02_bn_tanh_42545946034237
// MI455X (gfx1250) — compile-verified
//
#include <hip/hip_runtime.h>
#include <hip/hip_bf16.h>
#include <math.h>

typedef __attribute__((ext_vector_type(2))) float v2f;
typedef __attribute__((ext_vector_type(8))) float v8f;

#define N_NODES 50000
#define BN_EPS 1e-5f

// ---------------- zero fill (grid-stride) ----------------
__global__ void fill_zero(float* __restrict__ p, long long n) {
    long long i = (long long)blockIdx.x * blockDim.x + threadIdx.x;
    long long stride = (long long)gridDim.x * blockDim.x;
    for (; i < n; i += stride) p[i] = 0.0f;
}

// ---------------- edge scatter: agg[dst] += w_e * x[src] ----------------
// grid: (ceil(E/256), F/4), block: 256.  float4 gather, per-float atomic scatter.
__global__ void scatter_edges(const float* __restrict__ x,
                              const int* __restrict__ src,
                              const int* __restrict__ dst,
                              const float* __restrict__ w,
                              float* __restrict__ agg,
                              int F, int E) {
    int e = blockIdx.x * blockDim.x + threadIdx.x;
    if (e >= E) return;
    int c = blockIdx.y * 4;
    int s = src[e];
    int d = dst[e];
    float we = w[e];
    const float4 v = *(const float4*)(x + (size_t)s * F + c);
    float* out = agg + (size_t)d * F + c;
    atomicAdd(out + 0, we * v.x);
    atomicAdd(out + 1, we * v.y);
    atomicAdd(out + 2, we * v.z);
    atomicAdd(out + 3, we * v.w);
}

// ---------------- dual GEMM via fp32 WMMA ----------------
// out[N,Fout] = A0 @ Wrel^T + bias + A1 @ Wroot^T
// One wave (32 threads) per 16x16 output tile. grid: (N/16, Fout/16), block: 32.
// V_WMMA_F32_16X16X4_F32 layouts (ISA 7.12.2):
//   A (16x4): lane&15 = M row; lanes>=16 carry K=2,3; vgpr j -> K = 2*(lane>>4)+j
//   B (4x16): lane&15 = N col; same K mapping.  B[k][n] = W[n*K + k] (W is [Fout,K] row-major)
//   C/D     : lane&15 = N col; vgpr v -> M row = v + 8*(lane>>4)
__global__ void gemm_dual_wmma(const float* __restrict__ A0,   // agg [N,K]
                               const float* __restrict__ A1,   // x/h [N,K]
                               const float* __restrict__ Wrel, // [Fout,K]
                               const float* __restrict__ bias, // [Fout]
                               const float* __restrict__ Wroot,// [Fout,K]
                               float* __restrict__ out,        // [N,Fout]
                               int K, int Fout) {
    const int lane = threadIdx.x;
    const int half = lane >> 4;       // 0 or 1
    const int l16  = lane & 15;
    const int mBase = blockIdx.x * 16;
    const int nBase = blockIdx.y * 16;
    const int ncol  = nBase + l16;

    const float* a0 = A0 + (size_t)(mBase + l16) * K + 2 * half;
    const float* a1 = A1 + (size_t)(mBase + l16) * K + 2 * half;
    const float* b0 = Wrel  + (size_t)ncol * K + 2 * half;
    const float* b1 = Wroot + (size_t)ncol * K + 2 * half;

    v8f acc;
    const float bv = bias[ncol];
#pragma unroll
    for (int v = 0; v < 8; ++v) acc[v] = bv;

    for (int k = 0; k < K; k += 4) {
        v2f va0 = *(const v2f*)(a0 + k);
        v2f vb0 = *(const v2f*)(b0 + k);
        v2f va1 = *(const v2f*)(a1 + k);
        v2f vb1 = *(const v2f*)(b1 + k);
        acc = __builtin_amdgcn_wmma_f32_16x16x4_f32(false, va0, false, vb0,
                                                    (short)0, acc, false, false);
        acc = __builtin_amdgcn_wmma_f32_16x16x4_f32(false, va1, false, vb1,
                                                    (short)0, acc, false, false);
    }

    float* o = out + (size_t)(mBase + 8 * half) * Fout + ncol;
#pragma unroll
    for (int v = 0; v < 8; ++v)
        o[(size_t)v * Fout] = acc[v];
}

// ---------------- BN column stats (coalesced: thread = column) ----------------
// grid: ceil(N/nrows) blocks, blockDim = H
__global__ void bn_stats(const float* __restrict__ h,
                         float* __restrict__ sum, float* __restrict__ sumsq,
                         int H, int nrows) {
    int col = threadIdx.x;
    int r0 = blockIdx.x * nrows;
    int r1 = r0 + nrows;
    if (r1 > N_NODES) r1 = N_NODES;
    float s = 0.0f, sq = 0.0f;
    for (int r = r0; r < r1; ++r) {
        float v = h[(size_t)r * H + col];
        s += v;
        sq += v * v;
    }
    atomicAdd(&sum[col], s);
    atomicAdd(&sumsq[col], sq);
}

__global__ void bn_finalize(const float* __restrict__ sum,
                            const float* __restrict__ sumsq,
                            const float* __restrict__ gamma,
                            const float* __restrict__ beta,
                            float* __restrict__ scale,
                            float* __restrict__ shift, int H) {
    int c = blockIdx.x * blockDim.x + threadIdx.x;
    if (c >= H) return;
    const float invN = 1.0f / (float)N_NODES;
    float mean = sum[c] * invN;
    float var  = sumsq[c] * invN - mean * mean;
    float sc = gamma[c] * rsqrtf(var + BN_EPS);
    scale[c] = sc;
    shift[c] = beta[c] - mean * sc;
}

// in-place h = tanh(h*scale[col] + shift[col]), vectorized by 4
__global__ void bn_tanh(float* __restrict__ h,
                        const float* __restrict__ scale,
                        const float* __restrict__ shift,
                        int H, long long total4) {
    long long i = (long long)blockIdx.x * blockDim.x + threadIdx.x;
    if (i >= total4) return;
    long long base = i * 4;
    int col = (int)(base % (long long)H);
    float4 v = *(float4*)(h + base);
    v.x = tanhf(v.x * scale[col + 0] + shift[col + 0]);
    v.y = tanhf(v.y * scale[col + 1] + shift[col + 1]);
    v.z = tanhf(v.z * scale[col + 2] + shift[col + 2]);
    v.w = tanhf(v.w * scale[col + 3] + shift[col + 3]);
    *(float4*)(h + base) = v;
}

extern "C" void kernel_launch(void* const* d_in, const int* in_sizes, int n_in,
                              void* d_out, int out_size, void* d_ws, size_t ws_size,
                              hipStream_t stream) {
    const float* x      = (const float*)d_in[0];
    const int*   ei     = (const int*)d_in[1];   // [2,E]: row0=src, row1=dst
    const float* w      = (const float*)d_in[2];
    const float* W1rel  = (const float*)d_in[3];
    const float* b1     = (const float*)d_in[4];
    const float* W1root = (const float*)d_in[5];
    const float* W2rel  = (const float*)d_in[6];
    const float* b2     = (const float*)d_in[7];
    const float* W2root = (const float*)d_in[8];
    const float* W3rel  = (const float*)d_in[9];
    const float* b3     = (const float*)d_in[10];
    const float* W3root = (const float*)d_in[11];
    const float* g1     = (const float*)d_in[12];
    const float* be1    = (const float*)d_in[13];
    const float* g2     = (const float*)d_in[14];
    const float* be2    = (const float*)d_in[15];

    const int E = in_sizes[1] / 2;
    const int* src = ei;
    const int* dst = ei + E;

    // workspace layout (floats)
    float* ws    = (float*)d_ws;
    float* agg   = ws;                                  // N*256 (reused all layers)
    float* h1    = agg + (size_t)N_NODES * 256;         // N*128
    float* h2    = h1  + (size_t)N_NODES * 128;         // N*256
    float* stats = h2  + (size_t)N_NODES * 256;         // 4*256
    float* sums  = stats;
    float* sumsq = stats + 256;
    float* scale = stats + 512;
    float* shift = stats + 768;

    auto layer = [&](const float* in, int K, const float* Wrel, const float* bias,
                     const float* Wroot, float* outp, int Fout) {
        fill_zero<<<2048, 256, 0, stream>>>(agg, (long long)N_NODES * K);
        dim3 sg((E + 255) / 256, K / 4);
        scatter_edges<<<sg, 256, 0, stream>>>(in, src, dst, w, agg, K, E);
        dim3 gg(N_NODES / 16, Fout / 16);   // N_NODES = 3125*16 exactly, no tail
        gemm_dual_wmma<<<gg, 32, 0, stream>>>(agg, in, Wrel, bias, Wroot, outp, K, Fout);
    };

    auto bn = [&](float* h, int H, const float* gamma, const float* beta) {
        fill_zero<<<4, 128, 0, stream>>>(stats, 512);
        const int RPB = 100;
        bn_stats<<<(N_NODES + RPB - 1) / RPB, H, 0, stream>>>(h, sums, sumsq, H, RPB);
        bn_finalize<<<1, 256, 0, stream>>>(sums, sumsq, gamma, beta, scale, shift, H);
        long long t4 = (long long)N_NODES * H / 4;
        bn_tanh<<<(int)((t4 + 255) / 256), 256, 0, stream>>>(h, scale, shift, H, t4);
    };

    // Layer 1: 64 -> 128
    layer(x, 64, W1rel, b1, W1root, h1, 128);
    bn(h1, 128, g1, be1);
    // Layer 2: 128 -> 256
    layer(h1, 128, W2rel, b2, W2root, h2, 256);
    bn(h2, 256, g2, be2);
    // Layer 3: 256 -> 32, straight to output
    layer(h2, 256, W3rel, b3, W3root, (float*)d_out, 32);
}